// model_benchmark_split_58548994179555
// MI455X (gfx1250) — compile-verified
//
#include <hip/hip_runtime.h>

// CDNA5 / gfx1250, wave32. Single-workgroup, single-wave persistent kernel.
// All GEMMs via V_WMMA_F32_16X16X4_F32 (fp32 precision preserved).

typedef __attribute__((ext_vector_type(2))) float v2f;
typedef __attribute__((ext_vector_type(8))) float v8f;

#define LDA   36      // activation LDS row stride (floats): conflict-free on 64 banks
#define BUFN  (32*LDA)

// ---- LDS pool offsets (floats) ----
#define O_NI     0
#define O_X      (1*BUFN)
#define O_XC     (2*BUFN)
#define O_T0     (3*BUFN)
#define O_T1     (4*BUFN)
#define O_S      (5*BUFN)
#define O_SUMM   (6*BUFN)
#define O_DAG    (7*BUFN)
#define O_BACK   (8*BUFN)
#define O_EXT    (9*BUFN)
#define O_MERGE  (10*BUFN)
#define O_WBUF   (11*BUFN)          // 32x32 weight tile
#define O_BBUF   (O_WBUF + 1024)    // 32 bias
#define O_ZB     (O_BBUF + 32)      // 32 zero-bias
#define O_GLOB   (O_ZB + 32)        // 16
#define O_LOGIT  (O_GLOB + 16)      // 32
#define SMEM_TOT (O_LOGIT + 32)     // 13808 floats = 55,232 bytes

// One wave computes O[32 x 16*ntiles] = act(A[32 x 4*ktiles] * B + bias)
// using V_WMMA_F32_16X16X4_F32 tiles. A rows 20..31 are zero-padded garbage
// and only ever pollute output rows 20..31 (never read as K<=20 columns).
__device__ __forceinline__ void gemm_tiles(const float* A, int lda,
                                           const float* B, int ldb,
                                           const float* bias,
                                           float* O, int ldo,
                                           int ktiles, int ntiles,
                                           bool do_relu, int lane)
{
    const int l15   = lane & 15;
    const int khalf = (lane >> 4) << 1;   // 0 or 2 (A/B VGPR layout split)
    const int madd  = (lane >> 4) << 3;   // 0 or 8 (C/D VGPR layout split)
    for (int nt = 0; nt < ntiles; ++nt) {
        const int   ncol = nt * 16 + l15;
        const float bv   = bias[ncol];
        for (int mt = 0; mt < 2; ++mt) {
            const int mrow = mt * 16 + l15;
            v8f acc = {0.f, 0.f, 0.f, 0.f, 0.f, 0.f, 0.f, 0.f};
            for (int kt = 0; kt < ktiles; ++kt) {
                const int k0 = kt * 4 + khalf;
                v2f a, b;
                a.x = A[mrow * lda + k0];          // A 16x4 frag: lanes0-15 K=k0,k0+1
                a.y = A[mrow * lda + k0 + 1];      //              lanes16-31 K=k0+2,k0+3
                b.x = B[(k0    ) * ldb + ncol];    // B 4x16 frag: row K striped over lanes
                b.y = B[(k0 + 1) * ldb + ncol];
                acc = __builtin_amdgcn_wmma_f32_16x16x4_f32(
                        /*neg_a=*/false, a, /*neg_b=*/false, b,
                        /*c_mod=*/(short)0, acc, /*reuse_a=*/false, /*reuse_b=*/false);
            }
#pragma unroll
            for (int r = 0; r < 8; ++r) {          // D: VGPR r -> rows M=r / M=r+8
                float v = acc[r] + bv;
                if (do_relu) v = fmaxf(v, 0.0f);
                O[(mt * 16 + madd + r) * ldo + ncol] = v;
            }
        }
    }
}

// Load a KxN weight + N bias into zero-padded 32x32 / 32 LDS tiles.
__device__ __forceinline__ void load_wb(const float* W, int K, int N,
                                        const float* b,
                                        float* Wbuf, float* Bbuf, int lane)
{
    for (int i = lane; i < 1024; i += 32) Wbuf[i] = 0.0f;
    Bbuf[lane] = 0.0f;
    __syncthreads();
    for (int i = lane; i < K * N; i += 32) Wbuf[(i / N) * 32 + (i % N)] = W[i];
    if (lane < N) Bbuf[lane] = b[lane];
    __syncthreads();
}

__global__ __launch_bounds__(32)
void gnn_head_kernel(const float* __restrict__ inp,
                     const float* __restrict__ h1w, const float* __restrict__ h1b,
                     const float* __restrict__ h2w, const float* __restrict__ h2b,
                     const float* __restrict__ h3w, const float* __restrict__ h3b,
                     const float* __restrict__ d1w, const float* __restrict__ d1b,
                     const float* __restrict__ d2w, const float* __restrict__ d2b,
                     const float* __restrict__ d3w, const float* __restrict__ d3b,
                     const float* __restrict__ g1w, const float* __restrict__ g1b,
                     const float* __restrict__ g2w, const float* __restrict__ g2b,
                     const float* __restrict__ g3w, const float* __restrict__ g3b,
                     const float* __restrict__ f1w, const float* __restrict__ f1b,
                     const float* __restrict__ f2w, const float* __restrict__ f2b,
                     const float* __restrict__ f3w, const float* __restrict__ f3b,
                     const float* __restrict__ f4w, const float* __restrict__ f4b,
                     float* __restrict__ out)
{
    __shared__ float sm[SMEM_TOT];
    const int lane = threadIdx.x;   // blockDim.x == 32 (one wave)

    float* ni    = sm + O_NI;
    float* x     = sm + O_X;
    float* xc    = sm + O_XC;
    float* t0    = sm + O_T0;
    float* t1    = sm + O_T1;
    float* s     = sm + O_S;
    float* summ  = sm + O_SUMM;
    float* dag   = sm + O_DAG;
    float* back  = sm + O_BACK;
    float* ext   = sm + O_EXT;
    float* mrg   = sm + O_MERGE;
    float* Wbuf  = sm + O_WBUF;
    float* Bbuf  = sm + O_BBUF;
    float* zb    = sm + O_ZB;
    float* glob  = sm + O_GLOB;
    float* logit = sm + O_LOGIT;

    // ---- zero whole LDS pool (guarantees zero padding everywhere) ----
    for (int i = lane; i < SMEM_TOT; i += 32) sm[i] = 0.0f;
    __syncthreads();

    // ---- stage inputs: ni (20x5), summ (20x20), backmap (20x20) ----
    for (int i = lane; i < 100; i += 32) ni[(i / 5) * LDA + (i % 5)] = inp[i];
    for (int i = lane; i < 400; i += 32) {
        summ[(i / 20) * LDA + (i % 20)] = inp[3480 + i];
        back[(i / 20) * LDA + (i % 20)] = inp[3900 + i];
    }
    __syncthreads();

    // ---- h1 -> h2 -> h3 : gcn trunk (fblock loop is *= zeros -> exact no-op) ----
    load_wb(h1w, 5, 16, h1b, Wbuf, Bbuf, lane);
    gemm_tiles(ni, LDA, Wbuf, 32, Bbuf, t0, LDA, 2, 1, true, lane);  __syncthreads();
    load_wb(h2w, 16, 8, h2b, Wbuf, Bbuf, lane);
    gemm_tiles(t0, LDA, Wbuf, 32, Bbuf, t1, LDA, 4, 1, true, lane);  __syncthreads();
    load_wb(h3w, 8, 8, h3b, Wbuf, Bbuf, lane);
    gemm_tiles(t1, LDA, Wbuf, 32, Bbuf, x,  LDA, 2, 1, true, lane);  __syncthreads();
    // x == gcn_output (20x8)

    // ---- xc = [ni | x]  (20x13) ----
    for (int i = lane; i < 20 * 13; i += 32) {
        int r = i / 13, c = i % 13;
        xc[r * LDA + c] = (c < 5) ? ni[r * LDA + c] : x[r * LDA + (c - 5)];
    }
    __syncthreads();

    // ---- d1 -> d2 -> d3 ----
    load_wb(d1w, 13, 16, d1b, Wbuf, Bbuf, lane);
    gemm_tiles(xc, LDA, Wbuf, 32, Bbuf, t0, LDA, 4, 1, true, lane);  __syncthreads();
    load_wb(d2w, 16, 8, d2b, Wbuf, Bbuf, lane);
    gemm_tiles(t0, LDA, Wbuf, 32, Bbuf, t1, LDA, 4, 1, true, lane);  __syncthreads();
    load_wb(d3w, 8, 8, d3b, Wbuf, Bbuf, lane);
    gemm_tiles(t1, LDA, Wbuf, 32, Bbuf, s,  LDA, 2, 1, true, lane);  __syncthreads();

    // ---- dag_summary = summ(20x20) @ s(20x8)   (K=20 -> 5 ktiles exact) ----
    gemm_tiles(summ, LDA, s, LDA, zb, dag, LDA, 5, 1, false, lane);  __syncthreads();

    // ---- g1 -> g2 -> g3 ----
    load_wb(g1w, 8, 16, g1b, Wbuf, Bbuf, lane);
    gemm_tiles(dag, LDA, Wbuf, 32, Bbuf, t0, LDA, 2, 1, true, lane); __syncthreads();
    load_wb(g2w, 16, 8, g2b, Wbuf, Bbuf, lane);
    gemm_tiles(t0, LDA, Wbuf, 32, Bbuf, t1, LDA, 4, 1, true, lane);  __syncthreads();
    load_wb(g3w, 8, 8, g3b, Wbuf, Bbuf, lane);
    gemm_tiles(t1, LDA, Wbuf, 32, Bbuf, s,  LDA, 2, 1, true, lane);  __syncthreads();

    // ---- global_summary = running(1x20) @ s(20x8) ----
    if (lane < 8) {
        float acc = 0.0f;
        for (int j = 0; j < 20; ++j) acc += inp[3880 + j] * s[j * LDA + lane];
        glob[lane] = acc;
    }
    __syncthreads();

    // ---- dag_ext = backmap(20x20) @ dag_summary(20x8) ----
    gemm_tiles(back, LDA, dag, LDA, zb, ext, LDA, 5, 1, false, lane); __syncthreads();

    // ---- merge = [ni | gcn | dag_ext | glob_ext]  (20x29, cols 29..31 stay 0) ----
    for (int i = lane; i < 20 * 29; i += 32) {
        int r = i / 29, c = i % 29;
        float v;
        if      (c < 5)  v = ni[r * LDA + c];
        else if (c < 13) v = x [r * LDA + (c - 5)];
        else if (c < 21) v = ext[r * LDA + (c - 13)];
        else             v = glob[c - 21];
        mrg[r * LDA + c] = v;
    }
    __syncthreads();

    // ---- fc1 (29->32, 2 N-tiles) -> fc2 (32->16) -> fc3 (16->8) ----
    load_wb(f1w, 29, 32, f1b, Wbuf, Bbuf, lane);
    gemm_tiles(mrg, LDA, Wbuf, 32, Bbuf, t0, LDA, 8, 2, true, lane); __syncthreads();
    load_wb(f2w, 32, 16, f2b, Wbuf, Bbuf, lane);
    gemm_tiles(t0, LDA, Wbuf, 32, Bbuf, t1, LDA, 8, 1, true, lane);  __syncthreads();
    load_wb(f3w, 16, 8, f3b, Wbuf, Bbuf, lane);
    gemm_tiles(t1, LDA, Wbuf, 32, Bbuf, xc, LDA, 4, 1, true, lane);  __syncthreads();

    // ---- fc4 (8->1) + validity mask ----
    if (lane < 20) {
        float acc = f4b[0];
        for (int k = 0; k < 8; ++k) acc += xc[lane * LDA + k] * f4w[k];
        acc += (inp[100 + lane] - 1.0f) * 10000.0f;
        logit[lane] = acc;
    }
    __syncthreads();

    // ---- softmax over 20 nodes ----
    if (lane < 20) {
        float m = -3.402823466e38f;
        for (int j = 0; j < 20; ++j) m = fmaxf(m, logit[j]);
        float sum = 0.0f;
        for (int j = 0; j < 20; ++j) sum += __expf(logit[j] - m);
        out[lane] = __expf(logit[lane] - m) / sum;
    }
}

extern "C" void kernel_launch(void* const* d_in, const int* in_sizes, int n_in,
                              void* d_out, int out_size, void* d_ws, size_t ws_size,
                              hipStream_t stream)
{
    (void)in_sizes; (void)n_in; (void)out_size; (void)d_ws; (void)ws_size;
    const float* inp = (const float*)d_in[0];
    // setup_inputs order: input, then {h1,h2,h3,f1,f2,f3,d1,d2,d3,g1,g2,g3,fc1..fc4} x (w,b).
    // f1/f2/f3 (indices 7..12) are dead: their outputs are multiplied by zeros.
    const float* h1w = (const float*)d_in[1];  const float* h1b = (const float*)d_in[2];
    const float* h2w = (const float*)d_in[3];  const float* h2b = (const float*)d_in[4];
    const float* h3w = (const float*)d_in[5];  const float* h3b = (const float*)d_in[6];
    const float* d1w = (const float*)d_in[13]; const float* d1b = (const float*)d_in[14];
    const float* d2w = (const float*)d_in[15]; const float* d2b = (const float*)d_in[16];
    const float* d3w = (const float*)d_in[17]; const float* d3b = (const float*)d_in[18];
    const float* g1w = (const float*)d_in[19]; const float* g1b = (const float*)d_in[20];
    const float* g2w = (const float*)d_in[21]; const float* g2b = (const float*)d_in[22];
    const float* g3w = (const float*)d_in[23]; const float* g3b = (const float*)d_in[24];
    const float* f1w = (const float*)d_in[25]; const float* f1b = (const float*)d_in[26];
    const float* f2w = (const float*)d_in[27]; const float* f2b = (const float*)d_in[28];
    const float* f3w = (const float*)d_in[29]; const float* f3b = (const float*)d_in[30];
    const float* f4w = (const float*)d_in[31]; const float* f4b = (const float*)d_in[32];

    gnn_head_kernel<<<dim3(1), dim3(32), 0, stream>>>(
        inp,
        h1w, h1b, h2w, h2b, h3w, h3b,
        d1w, d1b, d2w, d2b, d3w, d3b,
        g1w, g1b, g2w, g2b, g3w, g3b,
        f1w, f1b, f2w, f2b, f3w, f3b, f4w, f4b,
        (float*)d_out);
}